// BOLD_Layer_75058848464969
// MI455X (gfx1250) — compile-verified
//
#include <hip/hip_runtime.h>
#include <stddef.h>

// Balloon-Windkessel BOLD model: 2000 independent 4-state ODEs x 10000 Euler steps.
// Latency-bound sequential scan (63 wave32s chip-wide, one wave per SIMD, so NO
// occupancy-based latency hiding -- every exposed wait lands on the critical path).
// Strategy:
//  - minimal dependent chain: hardware v_log_f32 / v_exp_f32 / v_rcp_f32 + FMAs;
//    v^3.125 and v^2.125 share one log2(v)
//  - z (node_history) software-pipelined depth 2: load z[t+2] at iter t, consume
//    z[t] -> s_wait_loadcnt has ~2 iterations of slack
//  - output stage skewed by one iteration: at iter t, ds_load row t-1 FIRST,
//    compute (covers LDS load latency), coalesced 128B global stores of row t-1,
//    then ds_store row t.  Same-wave LDS ops are in-order in HW (DScnt), so only
//    zero-cost wave_barrier scheduling fences are needed -- no __syncthreads, no
//    global-store completion waits in the loop
//  - wave-uniform fast path (unconditional stores) for full waves; masked variant
//    only for the single 16-region tail wave
//  - global_prefetch_b8 (WGP scope) 24 rows ahead

#define KAPPA_C        0.65f
#define GAMMAB_C       0.41f
#define INV_TAO_C      (1.0f / 0.98f)
#define INV_RO_C       (1.0f / 0.34f)
#define V0_C           0.02f
#define K1_C           2.38f
#define K2_C           2.0f
#define K3_C           0.48f
#define INV_ALPHA_C    3.125f         // 1/0.32
#define INV_ALPHA_M1_C 2.125f         // 1/0.32 - 1
#define LOG2_1MRO_C    (-0.59946207f) // log2(1 - 0.34) = log2(0.66)
#define PF_DIST        24

__device__ __forceinline__
void bold_step(float z, float dt,
               float& x, float& f, float& v, float& q, float& bold)
{
    // ---- critical dependent chain: hardware log2/exp2/rcp ----
    const float lv    = __builtin_amdgcn_logf(v);                    // v_log_f32
    const float v_ia  = __builtin_amdgcn_exp2f(INV_ALPHA_C * lv);    // v^(1/a)
    const float v_ia1 = __builtin_amdgcn_exp2f(INV_ALPHA_M1_C * lv); // v^(1/a-1)
    const float rf    = __builtin_amdgcn_rcpf(f);                    // 1/f
    const float e     = __builtin_amdgcn_exp2f(LOG2_1MRO_C * rf);    // (1-ro)^(1/f)

    const float dx = z - KAPPA_C * x - GAMMAB_C * (f - 1.0f);
    const float df = x;
    const float dv = (f - v_ia) * INV_TAO_C;
    const float dq = fmaf(f * INV_RO_C, (1.0f - e), -q * v_ia1) * INV_TAO_C;

    x = fmaf(dt, dx, x);
    f = fmaf(dt, df, f);
    v = fmaf(dt, dv, v);
    q = fmaf(dt, dq, q);

    // BOLD from the *updated* state (matches reference)
    const float rv = __builtin_amdgcn_rcpf(v);
    bold = V0_C * (K1_C * (1.0f - q)
                 + K2_C * (1.0f - q * rv)
                 + K3_C * (1.0f - v));
}

template <bool FULL>
__device__ __forceinline__
void scan_loop(const float* __restrict__ nhp,   // &node_history[0*R + rc]
               float*       __restrict__ rowp,  // &hist[0*rowStride + wave_r0*5]
               float*       __restrict__ sh,    // 160 floats, this wave's chunk
               int lane, int R, int T, size_t rowStride, int validDw,
               float dt, float& x, float& f, float& v, float& q)
{
    if (T <= 0) return;

    const float* __restrict__ pfp = nhp + (size_t)PF_DIST * R;
    const int pfT = T - PF_DIST;

    // z software pipeline, depth 2:  zc = z[t], zn = z[t+1]
    float zc = nhp[0];
    float zn = (T > 1) ? nhp[(size_t)R] : zc;
    const float* __restrict__ zload = nhp + 2 * (size_t)R;

    // ---- peel t = 0: compute + stage (nothing to drain yet) ----
    {
        float zfut = (2 < T) ? zload[0] : zn;
        zload += R;
        if (0 < pfT) __builtin_prefetch(pfp, 0, 3);   // global_prefetch, WGP scope
        pfp += R;

        float bold;
        bold_step(zc, dt, x, f, v, q, bold);

        float* p = sh + lane * 5;
        p[0] = x; p[1] = f; p[2] = v; p[3] = q; p[4] = bold;
        __builtin_amdgcn_wave_barrier();

        zc = zn; zn = zfut;
    }

    // ---- steady state: t = 1 .. T-1 ----
    for (int t = 1; t < T; ++t) {
        // issue z[t+2] load (consumed two iterations from now)
        float zfut = (t + 2 < T) ? zload[0] : zn;
        zload += R;
        if (t < pfT) __builtin_prefetch(pfp, 0, 3);
        pfp += R;

        // ds_loads of row t-1 issued EARLY; compute below covers their latency.
        // Unconditional: invalid lanes read staged garbage, filtered at store.
        const float s0 = sh[lane];
        const float s1 = sh[lane + 32];
        const float s2 = sh[lane + 64];
        const float s3 = sh[lane + 96];
        const float s4 = sh[lane + 128];
        __builtin_amdgcn_wave_barrier();   // keep loads early / ordered vs stores

        float bold;
        bold_step(zc, dt, x, f, v, q, bold);

        // drain row t-1: coalesced contiguous b32, 128B per wave per store
        if (FULL || (lane       < validDw)) rowp[lane      ] = s0;
        if (FULL || (lane + 32  < validDw)) rowp[lane +  32] = s1;
        if (FULL || (lane + 64  < validDw)) rowp[lane +  64] = s2;
        if (FULL || (lane + 96  < validDw)) rowp[lane +  96] = s3;
        if (FULL || (lane + 128 < validDw)) rowp[lane + 128] = s4;
        rowp += rowStride;
        __builtin_amdgcn_wave_barrier();   // ds_stores below must stay below loads

        // stage row t (20B/lane -> LDS transpose)
        float* p = sh + lane * 5;
        p[0] = x; p[1] = f; p[2] = v; p[3] = q; p[4] = bold;
        __builtin_amdgcn_wave_barrier();

        zc = zn; zn = zfut;
    }

    // ---- drain final row T-1 ----
    {
        const float s0 = sh[lane];
        const float s1 = sh[lane + 32];
        const float s2 = sh[lane + 64];
        const float s3 = sh[lane + 96];
        const float s4 = sh[lane + 128];
        if (FULL || (lane       < validDw)) rowp[lane      ] = s0;
        if (FULL || (lane + 32  < validDw)) rowp[lane +  32] = s1;
        if (FULL || (lane + 64  < validDw)) rowp[lane +  64] = s2;
        if (FULL || (lane + 96  < validDw)) rowp[lane +  96] = s3;
        if (FULL || (lane + 128 < validDw)) rowp[lane + 128] = s4;
    }
}

__global__ __launch_bounds__(32)
void bold_scan_kernel(const float* __restrict__ init_state,   // [R,4]
                      const float* __restrict__ node_history, // [T,R]
                      const int*   __restrict__ step_size_p,  // [1]
                      float* __restrict__ out,                // [R*4] ++ [T,R,5]
                      int R, int T)
{
    __shared__ float sh[5 * 32];

    const int lane    = threadIdx.x;          // wave32 lane
    const int wave_r0 = blockIdx.x * 32;
    const int r       = wave_r0 + lane;
    const int rc      = (r < R) ? r : (R - 1);

    const float dt = (float)step_size_p[0] * 0.001f;

    float4 s0 = *(const float4*)(init_state + (size_t)rc * 4);
    float x = s0.x, f = s0.y, v = s0.z, q = s0.w;

    float* __restrict__ hist = out + (size_t)R * 4;
    const size_t rowStride   = (size_t)R * 5;
    const int validDw        = (R - wave_r0) * 5;

    const float* nhp  = node_history + rc;
    float*       rowp = hist + (size_t)wave_r0 * 5;

    if (validDw >= 5 * 32) {
        scan_loop<true >(nhp, rowp, sh, lane, R, T, rowStride, validDw,
                         dt, x, f, v, q);
    } else {
        scan_loop<false>(nhp, rowp, sh, lane, R, T, rowStride, validDw,
                         dt, x, f, v, q);
    }

    if (r < R) {
        float4 o; o.x = x; o.y = f; o.z = v; o.w = q;
        *(float4*)(out + (size_t)r * 4) = o;   // 16B-aligned final state
    }
}

extern "C" void kernel_launch(void* const* d_in, const int* in_sizes, int n_in,
                              void* d_out, int out_size, void* d_ws, size_t ws_size,
                              hipStream_t stream)
{
    const float* init_state   = (const float*)d_in[0];  // [R,4]
    const float* node_history = (const float*)d_in[1];  // [T,R]
    const int*   step_size_p  = (const int*)d_in[2];    // scalar on device
    // d_in[3] = sim_len (redundant: T derived from shapes)

    const int R = in_sizes[0] / 4;                 // 2000
    const int T = in_sizes[1] / (R > 0 ? R : 1);   // 10000

    float* out = (float*)d_out;

    const int threads = 32;                  // one wave32 per block -> spread across WGPs
    const int blocks  = (R + threads - 1) / threads;

    bold_scan_kernel<<<blocks, threads, 0, stream>>>(init_state, node_history,
                                                     step_size_p, out, R, T);
}